// LSTM_20280835572275
// MI455X (gfx1250) — compile-verified
//
#include <hip/hip_runtime.h>

// LSTM: B=32, T=512, I=512, H=1024, O=256
#define B_ 32
#define T_ 512
#define I_ 512
#define H_ 1024
#define O_ 256
#define NBLK 128   // persistent blocks: one (mtile,jtile) pair each

typedef float v2f __attribute__((ext_vector_type(2)));
typedef float v4f __attribute__((ext_vector_type(4)));
typedef float v8f __attribute__((ext_vector_type(8)));

__device__ __forceinline__ v4f ld4(const float* p) { return *(const v4f*)p; }
__device__ __forceinline__ float sigmoidf_(float x) { return 1.0f / (1.0f + __expf(-x)); }

// Two K=4 WMMAs fed from one float4 per operand. K-slot mapping is a
// per-instruction bijection applied identically to A and B, so the dot product
// over K is unchanged: wmma1 slots get {k, k+1, k+4h.. }, wmma2 gets {k+2, ...}.
__device__ __forceinline__ v8f wmma2(v4f a4, v4f b4, v8f acc) {
    v2f a01 = __builtin_shufflevector(a4, a4, 0, 1);
    v2f a23 = __builtin_shufflevector(a4, a4, 2, 3);
    v2f b01 = __builtin_shufflevector(b4, b4, 0, 1);
    v2f b23 = __builtin_shufflevector(b4, b4, 2, 3);
    acc = __builtin_amdgcn_wmma_f32_16x16x4_f32(false, a01, false, b01,
                                                (short)0, acc, false, false);
    acc = __builtin_amdgcn_wmma_f32_16x16x4_f32(false, a23, false, b23,
                                                (short)0, acc, false, false);
    return acc;
}

// Persistent full-sequence LSTM scan.
// Grid: 128 blocks (2 mtiles x 64 jtiles), 512 threads = 16 waves:
//   wave = gate(4) x ksplit(4):
//   ks0: x[0:256), ks1: x[256:512), ks2: h[0:512), ks3: h[512:1024).
// Cell state c stays in LDS for the whole scan. h ping-pongs through global;
// steps separated by agent-scope arrival-counter grid barrier.
__global__ __launch_bounds__(512) void lstm_scan(
    const float* __restrict__ x,      // [B,T,I]
    const float* __restrict__ h0,     // [1,B,H]
    const float* __restrict__ c0,     // [1,B,H]
    const float* __restrict__ W_ih,   // [4H,I]
    const float* __restrict__ W_hh,   // [4H,H]
    const float* __restrict__ b_ih,   // [4H]
    const float* __restrict__ b_hh,   // [4H]
    float* __restrict__ hbuf0,        // [B,H]
    float* __restrict__ hbuf1,        // [B,H]
    float* __restrict__ cfin,         // [B,H] final cell state out
    float* __restrict__ hs,           // [B,T,H]
    int*   __restrict__ counters)     // [T] zeroed arrival counters
{
    __shared__ float partial[4][4][256];  // [gate][ksplit][16x16 tile]
    __shared__ float c_lds[256];
    __shared__ float bias_lds[64];        // [gate][16]

    const int tid  = threadIdx.x;
    const int lane = tid & 31;
    // wave-uniform values, made compiler-visibly uniform -> scalar control flow
    const int wvu   = __builtin_amdgcn_readfirstlane(tid >> 5);  // 0..15
    const int gateu = wvu >> 2;                                  // 0..3
    const int ksu   = wvu & 3;                                   // K slice
    const int m0   = (blockIdx.x & 1) * 16;
    const int j0   = (blockIdx.x >> 1) * 16;
    const int half = lane >> 4;
    const int lr   = lane & 15;
    const int n    = gateu * H_ + j0 + lr;   // W row feeding B-fragment lane

    // --- init block-resident cell state and biases ---
    if (tid < 256) {
        const int b = m0 + (tid >> 4);
        const int j = j0 + (tid & 15);
        c_lds[tid] = c0[(size_t)b * H_ + j];
    }
    if (tid < 64) {
        const int g  = tid >> 4;
        const int jj = tid & 15;
        bias_lds[tid] = b_ih[g * H_ + j0 + jj] + b_hh[g * H_ + j0 + jj];
    }
    __syncthreads();

    // per-wave invariant pointers (pre-offset by float4 K sub-slice select)
    const float* xrow    = x    + (size_t)(m0 + lr) * (T_ * I_) + 4 * half;
    const float* wih_row = W_ih + (size_t)n * I_ + 4 * half;
    const float* whh_row = W_hh + (size_t)n * H_ + 4 * half;
    const int kx = ksu * 256;        // ksu<2  : x K-range [kx, kx+256)
    const int kh = (ksu - 2) * 512;  // ksu>=2 : h K-range [kh, kh+512)

    for (int t = 0; t < T_; ++t) {
        const float* hin  = (t == 0) ? h0 : ((t & 1) ? hbuf1 : hbuf0);
        float*       hout = (t & 1) ? hbuf0 : hbuf1;

        v8f acc = {};
        if (ksu < 2) {
            const float* arow = xrow + (size_t)t * I_;
            for (int k = kx; k < kx + 256; k += 8)
                acc = wmma2(ld4(arow + k), ld4(wih_row + k), acc);
        } else {
            const float* arow = hin + (size_t)(m0 + lr) * H_ + 4 * half;
            for (int k = kh; k < kh + 512; k += 8)
                acc = wmma2(ld4(arow + k), ld4(whh_row + k), acc);
        }

        // C/D layout: acc[r] is element (M = r + half*8, N = lr)
        #pragma unroll
        for (int r = 0; r < 8; ++r)
            partial[gateu][ksu][(r + half * 8) * 16 + lr] = acc[r];
        __syncthreads();

        // --- fused elementwise LSTM update (waves 0..7) ---
        if (wvu < 8) {
            const int e = tid;   // 0..255
            const int N = e & 15;
            float gi = partial[0][0][e] + partial[0][1][e] + partial[0][2][e]
                     + partial[0][3][e] + bias_lds[0 * 16 + N];
            float gf = partial[1][0][e] + partial[1][1][e] + partial[1][2][e]
                     + partial[1][3][e] + bias_lds[1 * 16 + N];
            float gg = partial[2][0][e] + partial[2][1][e] + partial[2][2][e]
                     + partial[2][3][e] + bias_lds[2 * 16 + N];
            float go = partial[3][0][e] + partial[3][1][e] + partial[3][2][e]
                     + partial[3][3][e] + bias_lds[3 * 16 + N];
            gi = sigmoidf_(gi);
            gf = sigmoidf_(gf);
            gg = tanhf(gg);
            go = sigmoidf_(go);
            const float cn = gf * c_lds[e] + gi * gg;
            const float hn = go * tanhf(cn);
            c_lds[e] = cn;
            const int b = m0 + (e >> 4);
            const int j = j0 + N;
            hout[(size_t)b * H_ + j] = hn;
            hs[((size_t)b * T_ + t) * H_ + j] = hn;
            if (t == T_ - 1) cfin[(size_t)b * H_ + j] = cn;
        }

        // --- grid barrier between timesteps ---
        if (t < T_ - 1) {
            __syncthreads();   // all waves' stores issued & elementwise done
            if (tid == 0) {
                __builtin_amdgcn_fence(__ATOMIC_RELEASE, "agent");   // push h to L2
                __hip_atomic_fetch_add(&counters[t], 1, __ATOMIC_RELAXED,
                                       __HIP_MEMORY_SCOPE_AGENT);
                while (__hip_atomic_load(&counters[t], __ATOMIC_RELAXED,
                                         __HIP_MEMORY_SCOPE_AGENT) < NBLK) {
                    __builtin_amdgcn_s_sleep(1);
                }
                __builtin_amdgcn_fence(__ATOMIC_ACQUIRE, "agent");   // invalidate WGP$
            }
            __syncthreads();
        }
    }
}

// FC head: out = hs_flat @ fc_W^T + fc_b.  M=B*T=16384, N=O=256, K=H.
// 2x2 register tiles per wave (32x32 output) -> half the L2 traffic of 1x1,
// 4 WMMAs per 2 float4 loads. 4096 waves = 512 blocks x 8 waves.
__global__ __launch_bounds__(256) void fc_head(
    const float* __restrict__ hs,     // [B*T, H]
    const float* __restrict__ fc_W,   // [O, H]
    const float* __restrict__ fc_b,   // [O]
    float* __restrict__ out)          // [B*T, O]
{
    const int tid   = threadIdx.x;
    const int lane  = tid & 31;
    const int wglob = (int)((blockIdx.x * 256u + tid) >> 5);  // 0..4095
    const int n0    = (wglob & 7) * 32;
    const int m0    = (wglob >> 3) * 32;
    const int half  = lane >> 4;
    const int lr    = lane & 15;

    const float* arow0 = hs   + (size_t)(m0 + lr) * H_ + 4 * half;
    const float* arow1 = arow0 + (size_t)16 * H_;
    const float* brow0 = fc_W + (size_t)(n0 + lr) * H_ + 4 * half;
    const float* brow1 = brow0 + (size_t)16 * H_;

    v8f acc00 = {}, acc01 = {}, acc10 = {}, acc11 = {};
    for (int k = 0; k < H_; k += 8) {
        v4f a0 = ld4(arow0 + k);
        v4f a1 = ld4(arow1 + k);
        v4f b0 = ld4(brow0 + k);
        v4f b1 = ld4(brow1 + k);
        acc00 = wmma2(a0, b0, acc00);
        acc01 = wmma2(a0, b1, acc01);
        acc10 = wmma2(a1, b0, acc10);
        acc11 = wmma2(a1, b1, acc11);
    }

    const float bias0 = fc_b[n0 + lr];
    const float bias1 = fc_b[n0 + 16 + lr];
    #pragma unroll
    for (int r = 0; r < 8; ++r) {
        const int m = m0 + r + half * 8;
        out[(size_t)m * O_ + (n0 + lr)]            = acc00[r] + bias0;
        out[(size_t)m * O_ + (n0 + 16 + lr)]       = acc01[r] + bias1;
        out[(size_t)(m + 16) * O_ + (n0 + lr)]     = acc10[r] + bias0;
        out[(size_t)(m + 16) * O_ + (n0 + 16 + lr)] = acc11[r] + bias1;
    }
}

extern "C" void kernel_launch(void* const* d_in, const int* in_sizes, int n_in,
                              void* d_out, int out_size, void* d_ws, size_t ws_size,
                              hipStream_t stream) {
    const float* x    = (const float*)d_in[0];
    const float* h0   = (const float*)d_in[1];
    const float* c0   = (const float*)d_in[2];
    const float* W_ih = (const float*)d_in[3];
    const float* W_hh = (const float*)d_in[4];
    const float* b_ih = (const float*)d_in[5];
    const float* b_hh = (const float*)d_in[6];
    const float* fc_W = (const float*)d_in[7];
    const float* fc_b = (const float*)d_in[8];
    float* out = (float*)d_out;

    // workspace: hs [B,T,H] | hbuf0 [B,H] | hbuf1 [B,H] | cbuf [B,H] | counters [T]
    float* hs    = (float*)d_ws;
    float* hbuf0 = hs + (size_t)B_ * T_ * H_;
    float* hbuf1 = hbuf0 + (size_t)B_ * H_;
    float* cbuf  = hbuf1 + (size_t)B_ * H_;
    int*   ctr   = (int*)(cbuf + (size_t)B_ * H_);

    // zero the grid-barrier counters every call (deterministic, capture-safe)
    hipMemsetAsync(ctr, 0, T_ * sizeof(int), stream);

    // one persistent dispatch does the entire T=512 scan
    lstm_scan<<<dim3(NBLK), dim3(512), 0, stream>>>(
        x, h0, c0, W_ih, W_hh, b_ih, b_hh, hbuf0, hbuf1, cbuf, hs, ctr);

    // FC head: 4096 waves, 8 waves per 256-thread block -> 512 blocks
    fc_head<<<dim3(512), dim3(256), 0, stream>>>(hs, fc_W, fc_b, out);

    // tail outputs: hT then cT appended after [B*T,O] (final h landed in hbuf0)
    const size_t bh_bytes = (size_t)B_ * H_ * sizeof(float);
    float* tail = out + (size_t)B_ * T_ * O_;
    hipMemcpyAsync(tail,           hbuf0, bh_bytes, hipMemcpyDeviceToDevice, stream);
    hipMemcpyAsync(tail + B_ * H_, cbuf,  bh_bytes, hipMemcpyDeviceToDevice, stream);
}